// SAGEGraph_85203561218589
// MI455X (gfx1250) — compile-verified
//
#include <hip/hip_runtime.h>
#include <hip/hip_bf16.h>

#define N_NODES 50000
#define N_EDGES 800000
#define N_GRAPHS 64
#define IN_F 128
#define H_F 128
#define N_CLS 32

typedef __attribute__((ext_vector_type(2))) float v2f;
typedef __attribute__((ext_vector_type(8))) float v8f;

// ---------------------------------------------------------------------------
// Embedding lookup + max_norm renorm + node-weight scale.  One wave per node,
// each of 32 lanes handles 4 of the 128 features; norm via lane butterfly.
// ---------------------------------------------------------------------------
__global__ void k_embed(const int* __restrict__ attr,
                        const float* __restrict__ node_w,
                        const float* __restrict__ table,
                        float* __restrict__ h0) {
  int wid  = threadIdx.x >> 5;
  int lane = threadIdx.x & 31;
  int node = blockIdx.x * (blockDim.x >> 5) + wid;
  if (node >= N_NODES) return;
  const float* row = table + (size_t)attr[node] * IN_F;
  float4 v = *(const float4*)(row + lane * 4);
  float ss = v.x * v.x + v.y * v.y + v.z * v.z + v.w * v.w;
  #pragma unroll
  for (int off = 16; off > 0; off >>= 1) ss += __shfl_xor(ss, off, 32);
  float nrm   = sqrtf(ss);
  float scale = (nrm > 1.0f) ? (1.0f / (nrm + 1e-7f)) : 1.0f;
  scale *= node_w[node];
  float4 o = make_float4(v.x * scale, v.y * scale, v.z * scale, v.w * scale);
  *(float4*)(h0 + (size_t)node * IN_F + lane * 4) = o;
}

// ---------------------------------------------------------------------------
// In-degree (count of edges landing on each node).
// ---------------------------------------------------------------------------
__global__ void k_degree(const int* __restrict__ dst, float* __restrict__ deg) {
  int e = blockIdx.x * blockDim.x + threadIdx.x;
  if (e < N_EDGES) atomicAdd(&deg[dst[e]], 1.0f);
}

// inv_deg in place: deg>0 ? 1/deg : 0  (implements the where(deg>0, s/deg, 0))
__global__ void k_invdeg(float* __restrict__ deg) {
  int n = blockIdx.x * blockDim.x + threadIdx.x;
  if (n < N_NODES) {
    float d = deg[n];
    deg[n] = (d > 0.0f) ? (1.0f / d) : 0.0f;
  }
}

// ---------------------------------------------------------------------------
// Edge scatter: S[dst] += edge_w * h[src].  Thread t handles (edge, 4 feats).
// Reads of h[src] rows are vectorized (b128); writes are f32 L2 atomics.
// ---------------------------------------------------------------------------
__global__ void k_scatter(const int* __restrict__ src, const int* __restrict__ dst,
                          const float* __restrict__ ew,
                          const float* __restrict__ h, float* __restrict__ S) {
  long stride = (long)gridDim.x * blockDim.x;
  for (long t = blockIdx.x * (long)blockDim.x + threadIdx.x;
       t < (long)N_EDGES * 32; t += stride) {
    int e    = (int)(t >> 5);
    int lane = (int)(t & 31);
    int s = src[e], d = dst[e];
    float w = ew[e];
    float4 hv = *(const float4*)(h + (size_t)s * IN_F + lane * 4);
    float* o  = S + (size_t)d * IN_F + lane * 4;
    atomicAdd(o + 0, w * hv.x);
    atomicAdd(o + 1, w * hv.y);
    atomicAdd(o + 2, w * hv.z);
    atomicAdd(o + 3, w * hv.w);
  }
}

// ---------------------------------------------------------------------------
// Fused SAGE linear: Y = Xs @ Ws + (inv_deg .* Xn) @ Wn + bias
// One wave per 16x16 output tile, V_WMMA_F32_16X16X4_F32, K = 128+128.
// Weight slab (2 x 128 x 16 = 16KB) staged in LDS per workgroup (8 waves =
// 8 row tiles x 1 col tile).  Bias pre-loaded into the accumulator.
//
// Fragment layouts per CDNA5 ISA 7.12.2:
//   A (16x4 f32): lanes 0-15 -> M=lane, v0=K+0, v1=K+1 ; lanes 16-31 -> K+2,K+3
//   B (4x16 f32): lanes 0-15 -> N=lane, v0=K+0, v1=K+1 ; lanes 16-31 -> K+2,K+3
//   C/D (16x16) : VGPR r -> M=r (lanes 0-15) / M=8+r (lanes 16-31), N=lane&15
// ---------------------------------------------------------------------------
__global__ void k_gemm(const float* __restrict__ Xs, const float* __restrict__ Xn,
                       const float* __restrict__ inv,
                       const float* __restrict__ Ws, const float* __restrict__ Wn,
                       const float* __restrict__ bias,
                       float* __restrict__ Y, int ncols) {
  __shared__ float sWs[H_F * 16];
  __shared__ float sWn[H_F * 16];
  int tid = threadIdx.x;
  int n0  = blockIdx.y * 16;
  for (int idx = tid; idx < H_F * 16; idx += blockDim.x) {
    int k = idx >> 4, j = idx & 15;
    sWs[idx] = Ws[k * ncols + n0 + j];
    sWn[idx] = Wn[k * ncols + n0 + j];
  }
  __syncthreads();

  int wid  = tid >> 5;
  int lane = tid & 31;
  int tile = blockIdx.x * (blockDim.x >> 5) + wid;
  if (tile >= N_NODES / 16) return;   // uniform per wave: EXEC stays all-1s

  int m0    = tile * 16;
  int mr    = lane & 15;              // M (for A) / N (for B, C/D)
  int khalf = (lane >> 4) << 1;       // 0 or 2

  float bv = bias[n0 + mr];
  v8f c = {bv, bv, bv, bv, bv, bv, bv, bv};

  const float* arow_s = Xs + (size_t)(m0 + mr) * H_F;
  const float* arow_n = Xn + (size_t)(m0 + mr) * H_F;
  float sc = inv[m0 + mr];            // mean-aggregation fold-in

  #pragma unroll
  for (int k = 0; k < H_F; k += 4) {
    v2f a = *(const v2f*)(arow_s + k + khalf);
    v2f b;
    b.x = sWs[(k + khalf) * 16 + mr];
    b.y = sWs[(k + khalf + 1) * 16 + mr];
    c = __builtin_amdgcn_wmma_f32_16x16x4_f32(false, a, false, b,
                                              (short)0, c, false, false);
  }
  #pragma unroll
  for (int k = 0; k < H_F; k += 4) {
    v2f a = *(const v2f*)(arow_n + k + khalf);
    a.x *= sc; a.y *= sc;
    v2f b;
    b.x = sWn[(k + khalf) * 16 + mr];
    b.y = sWn[(k + khalf + 1) * 16 + mr];
    c = __builtin_amdgcn_wmma_f32_16x16x4_f32(false, a, false, b,
                                              (short)0, c, false, false);
  }

  int mup = (lane >> 4) << 3;         // 0 or 8
  #pragma unroll
  for (int r = 0; r < 8; r++)
    Y[(size_t)(m0 + r + mup) * ncols + n0 + mr] = c[r];
}

// ---------------------------------------------------------------------------
// Weighted per-graph mean pooling: LDS accumulation, one flush per block.
// One wave per node: 32 lanes == 32 classes.
// ---------------------------------------------------------------------------
__global__ void k_pool(const float* __restrict__ h2, const float* __restrict__ node_w,
                       const int* __restrict__ gid,
                       float* __restrict__ psum, float* __restrict__ pcnt) {
  __shared__ float ps[N_GRAPHS * N_CLS];
  __shared__ float pc[N_GRAPHS];
  for (int i = threadIdx.x; i < N_GRAPHS * N_CLS; i += blockDim.x) ps[i] = 0.0f;
  for (int i = threadIdx.x; i < N_GRAPHS; i += blockDim.x) pc[i] = 0.0f;
  __syncthreads();

  int lane  = threadIdx.x & 31;
  int wid   = threadIdx.x >> 5;
  int wgrid = gridDim.x * (blockDim.x >> 5);
  for (int node = blockIdx.x * (blockDim.x >> 5) + wid; node < N_NODES;
       node += wgrid) {
    int g   = gid[node];
    float w = node_w[node];
    float v = h2[(size_t)node * N_CLS + lane] * w;
    atomicAdd(&ps[g * N_CLS + lane], v);
    if (lane == 0) atomicAdd(&pc[g], 1.0f);
  }
  __syncthreads();
  for (int i = threadIdx.x; i < N_GRAPHS * N_CLS; i += blockDim.x)
    atomicAdd(&psum[i], ps[i]);
  for (int i = threadIdx.x; i < N_GRAPHS; i += blockDim.x)
    atomicAdd(&pcnt[i], pc[i]);
}

__global__ void k_finalize(const float* __restrict__ psum,
                           const float* __restrict__ pcnt,
                           float* __restrict__ out) {
  int i = blockIdx.x * blockDim.x + threadIdx.x;
  if (i < N_GRAPHS * N_CLS)
    out[i] = psum[i] / fmaxf(pcnt[i >> 5], 1.0f);  // N_CLS == 32
}

// ---------------------------------------------------------------------------
extern "C" void kernel_launch(void* const* d_in, const int* in_sizes, int n_in,
                              void* d_out, int out_size, void* d_ws, size_t ws_size,
                              hipStream_t stream) {
  (void)in_sizes; (void)n_in; (void)out_size; (void)ws_size;

  const int*   attr   = (const int*)d_in[0];
  const float* node_w = (const float*)d_in[1];
  const int*   src    = (const int*)d_in[2];
  const int*   dst    = (const int*)d_in[3];
  const float* edge_w = (const float*)d_in[4];
  const int*   gid    = (const int*)d_in[5];
  const float* table  = (const float*)d_in[6];
  const float* W1s    = (const float*)d_in[7];
  const float* W1n    = (const float*)d_in[8];
  const float* b1     = (const float*)d_in[9];
  const float* W2s    = (const float*)d_in[10];
  const float* W2n    = (const float*)d_in[11];
  const float* b2     = (const float*)d_in[12];
  float* out = (float*)d_out;

  // Workspace layout (floats)
  float* ws  = (float*)d_ws;
  float* h0  = ws;                                   // 50000*128
  float* S   = h0 + (size_t)N_NODES * IN_F;          // 50000*128 (neigh sums)
  float* h1  = S  + (size_t)N_NODES * IN_F;          // 50000*128
  float* h2  = h1 + (size_t)N_NODES * H_F;           // 50000*32
  float* deg = h2 + (size_t)N_NODES * N_CLS;         // 50000 (-> inv_deg)
  float* psum = deg + N_NODES;                       // 64*32
  float* pcnt = psum + N_GRAPHS * N_CLS;             // 64

  // Zero accumulation buffers (deg/psum/pcnt are contiguous).
  hipMemsetAsync(S, 0, (size_t)N_NODES * IN_F * sizeof(float), stream);
  hipMemsetAsync(deg, 0,
                 (size_t)(N_NODES + N_GRAPHS * N_CLS + N_GRAPHS) * sizeof(float),
                 stream);

  // 1) embedding + renorm + node-weight
  k_embed<<<(N_NODES + 7) / 8, 256, 0, stream>>>(attr, node_w, table, h0);
  // 2) degrees -> inv_deg
  k_degree<<<(N_EDGES + 255) / 256, 256, 0, stream>>>(dst, deg);
  k_invdeg<<<(N_NODES + 255) / 256, 256, 0, stream>>>(deg);

  // 3) layer 1: scatter then fused WMMA GEMM (128 -> 128)
  k_scatter<<<2048, 256, 0, stream>>>(src, dst, edge_w, h0, S);
  {
    dim3 grid((N_NODES / 16 + 7) / 8, H_F / 16);
    k_gemm<<<grid, 256, 0, stream>>>(h0, S, deg, W1s, W1n, b1, h1, H_F);
  }

  // 4) layer 2: re-zero, scatter, fused WMMA GEMM (128 -> 32)
  hipMemsetAsync(S, 0, (size_t)N_NODES * IN_F * sizeof(float), stream);
  k_scatter<<<2048, 256, 0, stream>>>(src, dst, edge_w, h1, S);
  {
    dim3 grid((N_NODES / 16 + 7) / 8, N_CLS / 16);
    k_gemm<<<grid, 256, 0, stream>>>(h1, S, deg, W2s, W2n, b2, h2, N_CLS);
  }

  // 5) weighted per-graph mean
  k_pool<<<512, 256, 0, stream>>>(h2, node_w, gid, psum, pcnt);
  k_finalize<<<(N_GRAPHS * N_CLS + 255) / 256, 256, 0, stream>>>(psum, pcnt, out);
}